// NMS_39187281609256
// MI455X (gfx1250) — compile-verified
//
#include <hip/hip_runtime.h>
#include <stdint.h>

// Problem constants (match reference)
constexpr int Bv = 8;
constexpr int Nv = 8192;
constexpr int Cv = 80;
constexpr int NUM_DET = 300;
constexpr float IOU_THR = 0.5f;
constexpr float SCORE_THR = 0.0f;
#define NEG_INF (-__builtin_inff())

// Output layout (all values stored as float in d_out, concatenated in return order)
constexpr int IDX_OFF = 0;                       // [B,300] box indices (as float)
constexpr int SC_OFF  = Bv * NUM_DET;            // [B,300] scores
constexpr int BX_OFF  = 2 * Bv * NUM_DET;        // [B,300,4] boxes
constexpr int CLS_OFF = BX_OFF + Bv * NUM_DET*4; // [B,300] classes (as float)
constexpr int LEN_OFF = CLS_OFF + Bv * NUM_DET;  // [B] lengths (as float)

// ---------------------------------------------------------------------------
// Kernel 0: transpose scores [B,N,C] -> [B,C,N] so per-class reads coalesce.
// ---------------------------------------------------------------------------
constexpr int TB = 64; // boxes per tile
__global__ __launch_bounds__(256)
void transpose_scores(const float* __restrict__ S, float* __restrict__ T) {
    __shared__ float tile[Cv][TB + 1];          // 80*65*4 = 20.8 KB, conflict-free
    const int nb = Nv / TB;                     // 128
    const int b  = blockIdx.x / nb;
    const int i0 = (blockIdx.x % nb) * TB;
    const int t  = threadIdx.x;
    const int TOT = TB * Cv;                    // 5120

    __builtin_prefetch(S + ((size_t)b * Nv + i0) * Cv + t, 0, 0);

    for (int e = t; e < TOT; e += 256) {
        int r = e / Cv, c = e % Cv;             // read coalesced over c
        tile[c][r] = S[((size_t)b * Nv + (i0 + r)) * Cv + c];
    }
    __syncthreads();
    for (int e = t; e < TOT; e += 256) {
        int c = e / TB, r = e % TB;             // write coalesced over r
        T[((size_t)(b * Cv + c)) * Nv + i0 + r] = tile[c][r];
    }
}

// ---------------------------------------------------------------------------
// Wave32 shuffle argmax reduction carrying the winner's box coordinates.
// ---------------------------------------------------------------------------
struct Best { float v; int i; float bx1, by1, bx2, by2; };

__device__ __forceinline__ void cmp_take(Best& a, const Best& b) {
    // tie-break: smaller index wins (matches argmax); -1 == 0xFFFFFFFF loses
    if (b.v > a.v || (b.v == a.v && (unsigned)b.i < (unsigned)a.i)) a = b;
}

__device__ __forceinline__ void wave_reduce(Best& c) {
#pragma unroll
    for (int off = 16; off > 0; off >>= 1) {
        Best o;
        o.v   = __shfl_down(c.v,   off);
        o.i   = __shfl_down(c.i,   off);
        o.bx1 = __shfl_down(c.bx1, off);
        o.by1 = __shfl_down(c.by1, off);
        o.bx2 = __shfl_down(c.bx2, off);
        o.by2 = __shfl_down(c.by2, off);
        cmp_take(c, o);
    }
}

// ---------------------------------------------------------------------------
// Kernel 1: per-(batch,class) greedy NMS. One block per (b,c); the class's
// 8192 scores + boxes live entirely in VGPRs (16 slots/lane, 512 lanes).
// Per iteration: wave shuffle reduce of the carried local best (2 barriers),
// then a single fused pass that applies IoU suppression AND computes the next
// iteration's thread-local argmax — one register scan per iteration, not two.
// ---------------------------------------------------------------------------
constexpr int K1T = 512;
constexpr int NWAVES = K1T / 32;                // 16
constexpr int SLOTS = Nv / K1T;                 // 16

__global__ __launch_bounds__(K1T)
void nms_per_class(const float* __restrict__ T, const float* __restrict__ boxes,
                   float* __restrict__ surv_sc, int* __restrict__ surv_idx,
                   int* __restrict__ counts) {
    __shared__ Best red[NWAVES];
    __shared__ Best s_win;

    const int bc = blockIdx.x;                  // b*Cv + c
    const int b  = bc / Cv;
    const int t  = threadIdx.x;
    const int w  = t >> 5;
    const int lane = t & 31;

    const float*  srow = T + (size_t)bc * Nv;
    const float4* brow = (const float4*)boxes + (size_t)b * Nv;

    __builtin_prefetch(srow + t, 0, 0);         // global_prefetch_b8
    __builtin_prefetch(brow + t, 0, 0);

    float sc[SLOTS];
    float x1[SLOTS], y1[SLOTS], x2[SLOTS], y2[SLOTS];

    // load + initial thread-local argmax (ascending i -> earliest max index)
    Best c; c.v = NEG_INF; c.i = -1;
    c.bx1 = 0.f; c.by1 = 0.f; c.bx2 = 0.f; c.by2 = 0.f;
#pragma unroll
    for (int k = 0; k < SLOTS; ++k) {
        const int i = k * K1T + t;
        sc[k] = srow[i];
        float4 bb = brow[i];
        x1[k] = bb.x; y1[k] = bb.y; x2[k] = bb.z; y2[k] = bb.w;
        if (sc[k] > c.v) {
            c.v = sc[k]; c.i = i;
            c.bx1 = x1[k]; c.by1 = y1[k]; c.bx2 = x2[k]; c.by2 = y2[k];
        }
    }

    int nsurv = 0;
    const int base = bc * NUM_DET;
    for (int it = 0; it < NUM_DET; ++it) {
        Best r = c;
        wave_reduce(r);                         // lane 0 of each wave holds partial
        if (lane == 0) red[w] = r;
        __syncthreads();
        if (t < 32) {
            Best d;
            if (t < NWAVES) d = red[t];
            else { d.v = NEG_INF; d.i = -1; d.bx1 = d.by1 = d.bx2 = d.by2 = 0.f; }
            wave_reduce(d);
            if (t == 0) s_win = d;
        }
        __syncthreads();
        const Best win = s_win;
        if (!(win.v > SCORE_THR)) break;        // class exhausted

        if (t == 0) { surv_sc[base + nsurv] = win.v; surv_idx[base + nsurv] = win.i; }

        // fused: suppress vs winner AND recompute local argmax in one pass
        const float aref = (win.bx2 - win.bx1) * (win.by2 - win.by1);
        c.v = NEG_INF; c.i = -1;
        c.bx1 = 0.f; c.by1 = 0.f; c.bx2 = 0.f; c.by2 = 0.f;
#pragma unroll
        for (int k = 0; k < SLOTS; ++k) {
            float ix1 = fmaxf(x1[k], win.bx1);
            float iy1 = fmaxf(y1[k], win.by1);
            float ix2 = fminf(x2[k], win.bx2);
            float iy2 = fminf(y2[k], win.by2);
            float inter = fmaxf(ix2 - ix1, 0.0f) * fmaxf(iy2 - iy1, 0.0f);
            float area  = (x2[k] - x1[k]) * (y2[k] - y1[k]);
            float iou   = inter / (area + aref - inter);
            if (iou > IOU_THR) sc[k] = NEG_INF; // suppress (includes winner, IoU==1)
            if (sc[k] > c.v) {
                c.v = sc[k]; c.i = k * K1T + t;
                c.bx1 = x1[k]; c.by1 = y1[k]; c.bx2 = x2[k]; c.by2 = y2[k];
            }
        }
        ++nsurv;
    }
    if (t == 0) counts[bc] = nsurv;
}

// ---------------------------------------------------------------------------
// Kernel 2: per-batch 80-way merge of score-descending survivor lists,
// emitting the global top-300 in order. Candidate scores (80*300 contiguous
// floats) staged into LDS via gfx1250 async-to-LDS copies (ASYNCcnt).
// Reduction: wave shuffle + 4-partial combine -> 2 barriers per step.
// ---------------------------------------------------------------------------
__global__ __launch_bounds__(128)
void merge_topk(const float* __restrict__ surv_sc, const int* __restrict__ surv_idx,
                const int* __restrict__ counts, const float* __restrict__ boxes,
                float* __restrict__ out) {
    extern __shared__ float s_cand[];           // Cv*NUM_DET = 24000 floats (96 KB)
    __shared__ int   heads[Cv];
    __shared__ float r2v[4];
    __shared__ int   r2i[4];
    __shared__ float s_bv;
    __shared__ int   s_bc;
    __shared__ int   s_len;

    const int b = blockIdx.x;
    const int t = threadIdx.x;
    const int w = t >> 5;
    const int lane = t & 31;

    // prefill inactive defaults for this batch
    for (int j = t; j < NUM_DET; j += 128) {
        out[IDX_OFF + b * NUM_DET + j] = -1.0f;
        out[SC_OFF  + b * NUM_DET + j] = 0.0f;
        out[CLS_OFF + b * NUM_DET + j] = -1.0f;
        float* bo = out + BX_OFF + ((size_t)b * NUM_DET + j) * 4;
        bo[0] = 0.0f; bo[1] = 0.0f; bo[2] = 0.0f; bo[3] = 0.0f;
    }
    if (t < Cv) heads[t] = 0;
    if (t == 0) s_len = 0;

    // async-stage this batch's candidate scores (contiguous region) into LDS
    const float* src = surv_sc + (size_t)b * Cv * NUM_DET;
    const int nvec = (Cv * NUM_DET) / 4;        // 6000 x 16B
    for (int v = t; v < nvec; v += 128) {
        unsigned lds = (unsigned)(uintptr_t)(&s_cand[v * 4]);
        unsigned long long g = (unsigned long long)(uintptr_t)(src + v * 4);
        asm volatile("global_load_async_to_lds_b128 %0, %1, off"
                     :: "v"(lds), "v"(g) : "memory");
    }
    asm volatile("s_wait_asynccnt 0" ::: "memory");
    __syncthreads();

    const int cbase = b * Cv;
    const int mycnt = (t < Cv) ? counts[cbase + t] : 0;

    for (int step = 0; step < NUM_DET; ++step) {
        float v = NEG_INF; int ci = -1;
        if (t < Cv) {
            int h = heads[t];
            if (h < mycnt) { v = s_cand[t * NUM_DET + h]; ci = t; }
        }
        // intra-wave shuffle argmax (tie -> smaller class id)
#pragma unroll
        for (int off = 16; off > 0; off >>= 1) {
            float v2 = __shfl_down(v, off);
            int   i2 = __shfl_down(ci, off);
            if (v2 > v || (v2 == v && (unsigned)i2 < (unsigned)ci)) { v = v2; ci = i2; }
        }
        if (lane == 0) { r2v[w] = v; r2i[w] = ci; }
        __syncthreads();
        if (t == 0) {
            float bv = r2v[0]; int bcc = r2i[0];
#pragma unroll
            for (int j = 1; j < 4; ++j) {
                float v2 = r2v[j]; int i2 = r2i[j];
                if (v2 > bv || (v2 == bv && (unsigned)i2 < (unsigned)bcc)) { bv = v2; bcc = i2; }
            }
            s_bv = bv; s_bc = bcc;
        }
        __syncthreads();
        const float bv = s_bv; const int bc2 = s_bc;
        if (!(bv > SCORE_THR)) break;           // all lists exhausted

        if (t == 0) {
            const int h  = heads[bc2]; heads[bc2] = h + 1;
            const int gi = surv_idx[(cbase + bc2) * NUM_DET + h];
            out[IDX_OFF + b * NUM_DET + step] = (float)gi;
            out[SC_OFF  + b * NUM_DET + step] = bv;
            out[CLS_OFF + b * NUM_DET + step] = (float)bc2;
            const float4 bb = ((const float4*)boxes)[(size_t)b * Nv + gi];
            float* bo = out + BX_OFF + ((size_t)b * NUM_DET + step) * 4;
            bo[0] = bb.x; bo[1] = bb.y; bo[2] = bb.z; bo[3] = bb.w;
            s_len = step + 1;
        }
        __syncthreads();                        // heads update visible next step
    }
    if (t == 0) out[LEN_OFF + b] = (float)s_len;
}

// ---------------------------------------------------------------------------
extern "C" void kernel_launch(void* const* d_in, const int* in_sizes, int n_in,
                              void* d_out, int out_size, void* d_ws, size_t ws_size,
                              hipStream_t stream) {
    (void)in_sizes; (void)n_in; (void)out_size; (void)ws_size;
    const float* scores = (const float*)d_in[0];   // [B,N,C]
    const float* boxes  = (const float*)d_in[1];   // [B,N,4]
    float* out = (float*)d_out;

    // workspace carve (~22.5 MB): T | surv_sc | surv_idx | counts
    float* T        = (float*)d_ws;
    float* surv_sc  = T + (size_t)Bv * Cv * Nv;
    int*   surv_idx = (int*)(surv_sc + (size_t)Bv * Cv * NUM_DET);
    int*   counts   = surv_idx + (size_t)Bv * Cv * NUM_DET;

    transpose_scores<<<Bv * (Nv / TB), 256, 0, stream>>>(scores, T);
    nms_per_class<<<Bv * Cv, K1T, 0, stream>>>(T, boxes, surv_sc, surv_idx, counts);
    merge_topk<<<Bv, 128, Cv * NUM_DET * sizeof(float), stream>>>(
        surv_sc, surv_idx, counts, boxes, out);
}